// GraphMAELoss_40346922778986
// MI455X (gfx1250) — compile-verified
//
#include <hip/hip_runtime.h>
#include <hip/hip_bf16.h>

typedef __attribute__((ext_vector_type(2))) float v2f;
typedef __attribute__((ext_vector_type(8))) float v8f;

#define NUM_GRAPHS 1024
#define FDIM 8
#define NPT 16   // contiguous nodes per thread (batch is sorted -> ~1 atomic flush/thread)

// --- zero the per-graph accumulators (seg_sum[1024] ++ seg_cnt[1024]) ---
__global__ void GraphMAE_zero_ws(float* __restrict__ ws) {
    int i = blockIdx.x * blockDim.x + threadIdx.x;
    if (i < 2 * NUM_GRAPHS) ws[i] = 0.0f;
}

// --- streaming pass: per-node L1 error, run-accumulated per sorted graph id ---
__global__ void GraphMAE_partial(const float* __restrict__ pred,
                                 const float* __restrict__ targ,
                                 const int*   __restrict__ batch,
                                 float* __restrict__ seg_sum,
                                 float* __restrict__ seg_cnt,
                                 int N) {
    long long t     = (long long)blockIdx.x * blockDim.x + threadIdx.x;
    long long start = t * NPT;
    if (start >= N) return;
    long long end = start + NPT;
    if (end > N) end = N;

    int   curg = batch[start];
    float rs = 0.0f, rc = 0.0f;

    for (long long i = start; i < end; ++i) {
        int g = batch[i];
        const float4* p = (const float4*)(pred + i * FDIM);  // 32B aligned -> b128
        const float4* q = (const float4*)(targ + i * FDIM);
        float4 p0 = p[0], p1 = p[1];
        float4 q0 = q[0], q1 = q[1];
        float e = fabsf(p0.x - q0.x) + fabsf(p0.y - q0.y) +
                  fabsf(p0.z - q0.z) + fabsf(p0.w - q0.w) +
                  fabsf(p1.x - q1.x) + fabsf(p1.y - q1.y) +
                  fabsf(p1.z - q1.z) + fabsf(p1.w - q1.w);
        if (g != curg) {                      // rare: graph boundary inside run
            atomicAdd(&seg_sum[curg], rs);
            atomicAdd(&seg_cnt[curg], rc);
            curg = g; rs = 0.0f; rc = 0.0f;
        }
        rs += e; rc += 1.0f;
    }
    atomicAdd(&seg_sum[curg], rs);
    atomicAdd(&seg_cnt[curg], rc);
}

// --- finalize: one full wave32; WMMA f32 16x16x4 ones-matmul as cross-lane reduce ---
__global__ void GraphMAE_finalize(const float* __restrict__ seg_sum,
                                  const float* __restrict__ seg_cnt,
                                  float* __restrict__ out) {
    int lane = threadIdx.x;            // exactly 32 threads -> EXEC all 1s for WMMA
    float p = 0.0f;
    for (int g = lane; g < NUM_GRAPHS; g += 32)
        p += seg_sum[g] / (seg_cnt[g] * (float)FDIM);

    // A (16x4 f32): VGPR0 holds lanes' partials -> A[m][0]=p[m], A[m][2]=p[m+16]; VGPR1=0.
    // B (4x16) all ones => D[m][n] = p[m] + p[m+16].
    v2f A; A.x = p;    A.y = 0.0f;
    v2f B; B.x = 1.0f; B.y = 1.0f;
    v8f C = {};
    v8f D = __builtin_amdgcn_wmma_f32_16x16x4_f32(
        /*neg_a=*/false, A, /*neg_b=*/false, B,
        /*c_mod=*/(short)0, C, /*reuse_a=*/false, /*reuse_b=*/false);

    // D vgpr r: lanes 0-15 -> rowsum[r], lanes 16-31 -> rowsum[8+r].
    float v = D[0] + D[1] + D[2] + D[3] + D[4] + D[5] + D[6] + D[7];
    // total over 32 lanes = v(lane 0) + v(lane 16)
    int   vi = __float_as_int(v);
    float lo = __int_as_float(__builtin_amdgcn_readlane(vi, 0));
    float hi = __int_as_float(__builtin_amdgcn_readlane(vi, 16));
    if (lane == 0)
        out[0] = (lo + hi) * (10000.0f / (float)NUM_GRAPHS);
}

extern "C" void kernel_launch(void* const* d_in, const int* in_sizes, int n_in,
                              void* d_out, int out_size, void* d_ws, size_t ws_size,
                              hipStream_t stream) {
    const float* pred  = (const float*)d_in[0];
    const float* targ  = (const float*)d_in[1];
    const int*   batch = (const int*)  d_in[2];
    // d_in[3] (x) is unused by the reference -> never read it (saves 128 MB of traffic).
    int N = in_sizes[2];

    float* seg_sum = (float*)d_ws;
    float* seg_cnt = seg_sum + NUM_GRAPHS;

    GraphMAE_zero_ws<<<(2 * NUM_GRAPHS + 255) / 256, 256, 0, stream>>>(seg_sum);

    long long nthreads = ((long long)N + NPT - 1) / NPT;
    int nblocks = (int)((nthreads + 255) / 256);
    GraphMAE_partial<<<nblocks, 256, 0, stream>>>(pred, targ, batch,
                                                  seg_sum, seg_cnt, N);

    GraphMAE_finalize<<<1, 32, 0, stream>>>(seg_sum, seg_cnt, (float*)d_out);
}